// S4Layer_11811160064012
// MI455X (gfx1250) — compile-verified
//
#include <hip/hip_runtime.h>
#include <hip/hip_bf16.h>
#include <math.h>
#include <stdint.h>

// ---- problem constants (match reference) ----
#define BB 16
#define LL 4096
#define DIN 64
#define HH 256
#define NN 32
#define DOUT 10
#define LN_EPS 1e-5f
#define KP 64          // K-panel staged in LDS (KP x 256 floats = 64 KB)

typedef __attribute__((ext_vector_type(2))) float v2f;
typedef __attribute__((ext_vector_type(8))) float v8f;
typedef int v4i_gcc __attribute__((vector_size(16)));   // matches builtin proto

#if defined(__has_builtin)
#if __has_builtin(__builtin_amdgcn_global_load_async_to_lds_b128)
#define HAVE_ASYNC_LDS 1
#endif
#endif

// generic->addrspace casts via integer round-trip (global VA unchanged;
// LDS generic pointer's low 32 bits are the LDS byte offset per ISA 10.2)
__device__ __forceinline__ void async_copy16(const float* g, float* l) {
#if defined(HAVE_ASYNC_LDS)
    __builtin_amdgcn_global_load_async_to_lds_b128(
        (__attribute__((address_space(1))) v4i_gcc*)(uintptr_t)g,
        (__attribute__((address_space(3))) v4i_gcc*)(uint32_t)(uintptr_t)l,
        0, 0);
#else
    l[0] = g[0]; l[1] = g[1]; l[2] = g[2]; l[3] = g[3];
#endif
}

__device__ __forceinline__ void async_fence() {
#if defined(HAVE_ASYNC_LDS)
#if __has_builtin(__builtin_amdgcn_s_wait_asynccnt)
    __builtin_amdgcn_s_wait_asynccnt(0);
#else
    asm volatile("s_wait_asynccnt 0" ::: "memory");
#endif
#endif
}

// =====================================================================
// 1) S4D discretization: lambda = exp(dt*A), C2 = 2*C*B*(lambda-1)/A
// =====================================================================
__global__ void s4d_params_kernel(const float* __restrict__ log_dt,
                                  const float* __restrict__ log_A_real,
                                  const float* __restrict__ A_imag,
                                  const float* __restrict__ B_re,
                                  const float* __restrict__ B_im,
                                  const float* __restrict__ C_re,
                                  const float* __restrict__ C_im,
                                  float* __restrict__ lam_re,
                                  float* __restrict__ lam_im,
                                  float* __restrict__ c2_re,
                                  float* __restrict__ c2_im) {
    int idx = blockIdx.x * blockDim.x + threadIdx.x;
    if (idx >= HH * NN) return;
    int h = idx / NN;
    float dt  = expf(log_dt[h]);
    float are = -expf(log_A_real[idx]);
    float aim = A_imag[idx];
    float mag = expf(dt * are);
    float sa, ca;
    sincosf(dt * aim, &sa, &ca);
    float lre = mag * ca;
    float lim = mag * sa;
    lam_re[idx] = lre;
    lam_im[idx] = lim;
    float er = lre - 1.0f;
    float ei = lim;
    float inv = 1.0f / (are * are + aim * aim);
    float qre = (er * are + ei * aim) * inv;
    float qim = (ei * are - er * aim) * inv;
    float bre = B_re[idx], bim = B_im[idx];
    float tre = qre * bre - qim * bim;
    float tim = qre * bim + qim * bre;
    float cre = C_re[idx], cim = C_im[idx];
    c2_re[idx] = 2.0f * (tre * cre - tim * cim);
    c2_im[idx] = 2.0f * (tre * cim + tim * cre);
}

// =====================================================================
// 2) fp32 WMMA GEMM, N fixed = 256, K multiple of 64.
//    Block: 256 threads = 8 waves in a 2(M) x 4(N) grid; block output
//    tile 64 x 256. Weights staged per K-panel into LDS via async
//    global->LDS b128 copies; each wave computes 32x64 with 8
//    v_wmma_f32_16x16x4_f32 accumulators.
// =====================================================================
__global__ void __launch_bounds__(256)
wmma_gemm_f32_kernel(const float* __restrict__ A,
                     const float* __restrict__ Bm,
                     const float* __restrict__ bias,
                     const float* __restrict__ resid,
                     float* __restrict__ C,
                     int M, int K) {
    constexpr int N = HH;                       // 256
    __shared__ float bpanel[KP * N];            // 64 KB

    int w    = threadIdx.x >> 5;                // wave 0..7
    int lane = threadIdx.x & 31;
    int wcol = w & 3;                           // N group (64 cols each)
    int wrow = w >> 2;                          // M group (32 rows each)

    int row0 = blockIdx.x * 64 + wrow * 32;     // wave covers rows [row0, row0+32)
    int col0 = wcol * 64;

    int m    = lane & 15;
    int n    = lane & 15;
    int half = lane >> 4;
    int kb   = half * 2;

    v8f acc[2][4];
#pragma unroll
    for (int r = 0; r < 2; ++r)
#pragma unroll
        for (int t = 0; t < 4; ++t) acc[r][t] = (v8f){0,0,0,0,0,0,0,0};

    const float* ar0 = A + (size_t)(row0 + m) * K;
    const float* ar1 = ar0 + (size_t)16 * K;

    for (int kp0 = 0; kp0 < K; kp0 += KP) {
        // ---- stage KP x 256 weight panel into LDS (contiguous) ----
        {
            const float* gsrc = Bm + (size_t)kp0 * N;
#pragma unroll
            for (int i = 0; i < (KP * N / 4) / 256; ++i) {
                int f4 = threadIdx.x + i * 256;        // float4 index
                async_copy16(gsrc + f4 * 4, bpanel + f4 * 4);
            }
            async_fence();
        }
        __syncthreads();

        // ---- WMMA over the panel ----
        for (int kk = 0; kk < KP; kk += 4) {
            int kg = kp0 + kk + kb;
            v2f a0, a1;
            a0.x = ar0[kg];     a0.y = ar0[kg + 1];
            a1.x = ar1[kg];     a1.y = ar1[kg + 1];
            const float* bp = &bpanel[(kk + kb) * N + col0 + n];
#pragma unroll
            for (int t = 0; t < 4; ++t) {
                v2f b;
                b.x = bp[16 * t];
                b.y = bp[16 * t + N];
                acc[0][t] = __builtin_amdgcn_wmma_f32_16x16x4_f32(
                    false, a0, false, b, (short)0, acc[0][t], false, false);
                acc[1][t] = __builtin_amdgcn_wmma_f32_16x16x4_f32(
                    false, a1, false, b, (short)0, acc[1][t], false, false);
            }
        }
        __syncthreads();
    }

#pragma unroll
    for (int r = 0; r < 2; ++r) {
#pragma unroll
        for (int t = 0; t < 4; ++t) {
#pragma unroll
            for (int j = 0; j < 8; ++j) {
                int row = row0 + 16 * r + j + 8 * half;
                int col = col0 + 16 * t + n;
                float v = acc[r][t][j];
                if (bias)  v += bias[col];
                if (resid) v += resid[(size_t)row * N + col];
                C[(size_t)row * N + col] = v;
            }
        }
    }
}

// =====================================================================
// 3) S4D diagonal scan: one wave per (b,h); lane = complex mode n.
// =====================================================================
__global__ void s4_scan_kernel(const float* __restrict__ u,
                               const float* __restrict__ lam_re,
                               const float* __restrict__ lam_im,
                               const float* __restrict__ c2_re,
                               const float* __restrict__ c2_im,
                               float* __restrict__ yconv) {
    int tid  = blockIdx.x * blockDim.x + threadIdx.x;
    int wid  = tid >> 5;
    int lane = threadIdx.x & 31;
    if (wid >= BB * HH) return;
    int b = wid / HH;
    int h = wid % HH;

    float lre = lam_re[h * NN + lane];
    float lim = lam_im[h * NN + lane];
    float c2r = c2_re[h * NN + lane];
    float c2i = c2_im[h * NN + lane];

    const float* up = u + ((size_t)b * LL) * HH + h;
    float* yp = yconv + ((size_t)b * LL) * HH + h;

    float sre = 0.0f, sim = 0.0f;
    for (int l = 0; l < LL; ++l) {
        float uv = up[(size_t)l * HH];
        float nre = fmaf(lre, sre, fmaf(-lim, sim, uv));
        float nim = fmaf(lre, sim, lim * sre);
        sre = nre; sim = nim;
        float contrib = fmaf(c2r, sre, -c2i * sim);
#pragma unroll
        for (int off = 16; off > 0; off >>= 1)
            contrib += __shfl_xor(contrib, off, 32);
        if (lane == 0) yp[(size_t)l * HH] = contrib;
    }
}

// =====================================================================
// 4) Elementwise: yact = gelu_erf(yconv + D[h]*u)
// =====================================================================
__global__ void gelu_skip_kernel(const float* __restrict__ yconv,
                                 const float* __restrict__ u,
                                 const float* __restrict__ Dv,
                                 float* __restrict__ yact) {
    size_t idx = (size_t)blockIdx.x * blockDim.x + threadIdx.x;
    if (idx >= (size_t)BB * LL * HH) return;
    int h = (int)(idx & (HH - 1));
    float y = yconv[idx] + Dv[h] * u[idx];
    yact[idx] = 0.5f * y * (1.0f + erff(y * 0.70710678118654752f));
}

// =====================================================================
// 5) LayerNorm + decoder: one wave per (b,l) row.
// =====================================================================
__global__ void ln_dec_kernel(const float* __restrict__ hres,
                              const float* __restrict__ ln_g,
                              const float* __restrict__ ln_b,
                              const float* __restrict__ Wd,
                              const float* __restrict__ bd,
                              float* __restrict__ out) {
    int tid  = blockIdx.x * blockDim.x + threadIdx.x;
    int row  = tid >> 5;
    int lane = threadIdx.x & 31;
    if (row >= BB * LL) return;

    const float* hp = hres + (size_t)row * HH;
    float v[HH / 32];
    float sum = 0.0f, sq = 0.0f;
#pragma unroll
    for (int j = 0; j < HH / 32; ++j) {
        float x = hp[lane + 32 * j];
        v[j] = x;
        sum += x;
        sq  += x * x;
    }
#pragma unroll
    for (int off = 16; off > 0; off >>= 1) {
        sum += __shfl_xor(sum, off, 32);
        sq  += __shfl_xor(sq, off, 32);
    }
    float mean = sum * (1.0f / HH);
    float var  = sq * (1.0f / HH) - mean * mean;
    float inv  = 1.0f / sqrtf(var + LN_EPS);
#pragma unroll
    for (int j = 0; j < HH / 32; ++j) {
        int c = lane + 32 * j;
        v[j] = (v[j] - mean) * inv * ln_g[c] + ln_b[c];
    }
#pragma unroll
    for (int d = 0; d < DOUT; ++d) {
        float acc = 0.0f;
#pragma unroll
        for (int j = 0; j < HH / 32; ++j)
            acc = fmaf(v[j], Wd[(lane + 32 * j) * DOUT + d], acc);
#pragma unroll
        for (int off = 16; off > 0; off >>= 1)
            acc += __shfl_xor(acc, off, 32);
        if (lane == 0) out[(size_t)row * DOUT + d] = acc + bd[d];
    }
}

// =====================================================================
// host launcher
// =====================================================================
extern "C" void kernel_launch(void* const* d_in, const int* in_sizes, int n_in,
                              void* d_out, int out_size, void* d_ws, size_t ws_size,
                              hipStream_t stream) {
    const float* x        = (const float*)d_in[0];
    const float* W_enc    = (const float*)d_in[1];
    const float* b_enc    = (const float*)d_in[2];
    const float* log_dt   = (const float*)d_in[3];
    const float* log_Ar   = (const float*)d_in[4];
    const float* A_im     = (const float*)d_in[5];
    const float* B_re     = (const float*)d_in[6];
    const float* B_im     = (const float*)d_in[7];
    const float* C_re     = (const float*)d_in[8];
    const float* C_im     = (const float*)d_in[9];
    const float* Dv       = (const float*)d_in[10];
    const float* W_out    = (const float*)d_in[11];
    const float* b_out    = (const float*)d_in[12];
    const float* ln_g     = (const float*)d_in[13];
    const float* ln_b     = (const float*)d_in[14];
    const float* W_dec    = (const float*)d_in[15];
    const float* b_dec    = (const float*)d_in[16];
    float* out = (float*)d_out;

    const size_t BLH = (size_t)BB * LL * HH * sizeof(float);   // 64 MB
    char* ws = (char*)d_ws;
    float* u      = (float*)(ws);
    float* yconv  = (float*)(ws + BLH);            // reused in-place as yact
    float* hres   = (float*)(ws + 2 * BLH);
    float* lam_re = (float*)(ws + 3 * BLH);
    float* lam_im = lam_re + HH * NN;
    float* c2_re  = lam_im + HH * NN;
    float* c2_im  = c2_re + HH * NN;

    // 1) discretize S4D parameters
    s4d_params_kernel<<<(HH * NN + 255) / 256, 256, 0, stream>>>(
        log_dt, log_Ar, A_im, B_re, B_im, C_re, C_im,
        lam_re, lam_im, c2_re, c2_im);

    // 2) encoder: u = x @ W_enc + b_enc   (M=65536, K=64)
    {
        int M = BB * LL;
        wmma_gemm_f32_kernel<<<M / 64, 256, 0, stream>>>(
            x, W_enc, b_enc, nullptr, u, M, DIN);
    }

    // 3) diagonal SSM scan: yconv = conv(k, u)
    {
        int waves = BB * HH;
        s4_scan_kernel<<<waves * 32 / 256, 256, 0, stream>>>(
            u, lam_re, lam_im, c2_re, c2_im, yconv);
    }

    // 4) yact = gelu(yconv + D*u), in place over yconv
    {
        size_t n = (size_t)BB * LL * HH;
        gelu_skip_kernel<<<(unsigned)((n + 255) / 256), 256, 0, stream>>>(
            yconv, u, Dv, yconv);
    }

    // 5) hres = yact @ W_out + b_out + u   (M=65536, K=256)
    {
        int M = BB * LL;
        wmma_gemm_f32_kernel<<<M / 64, 256, 0, stream>>>(
            yconv, W_out, b_out, u, hres, M, HH);
    }

    // 6) layernorm + decoder
    {
        int waves = BB * LL;
        ln_dec_kernel<<<waves * 32 / 256, 256, 0, stream>>>(
            hres, ln_g, ln_b, W_dec, b_dec, out);
    }
}